// EGNN_43353399885953
// MI455X (gfx1250) — compile-verified
//
#include <hip/hip_runtime.h>
#include <hip/hip_bf16.h>
#include <math.h>

#define N_NODES  50000
#define N_EDGES  800000
#define HID      64
#define DEPTH    4
#define EDGE_IN  145
#define KPAD     160
#define FEAT_IN  128
#define EPS_F    1e-8f

typedef __attribute__((ext_vector_type(2))) float v2f;
typedef __attribute__((ext_vector_type(8))) float v8f;

__device__ __forceinline__ float silu_f(float v) { return v / (1.0f + __expf(-v)); }

__device__ __forceinline__ v8f wmma4(v2f a, v2f b, v8f c) {
  // D = A(16x4,f32) * B(4x16,f32) + C(16x16,f32)
  return __builtin_amdgcn_wmma_f32_16x16x4_f32(false, a, false, b, (short)0, c, false, false);
}

// D[MT*16 x 64] = A[MT*16 x K] (LDS, row-major, stride astride) * W[K x 64] (global) + bias
template<int MT>
__device__ __forceinline__ void wave_gemm(const float* Alds, int astride,
                                          const float* __restrict__ W,
                                          const float* __restrict__ bias,
                                          int K, int lane, v8f acc[MT][4]) {
  const int lm = lane & 15;
  const int lh = lane >> 4;
#pragma unroll
  for (int nb = 0; nb < 4; ++nb) {
    float bv = bias[nb * 16 + lm];
#pragma unroll
    for (int mt = 0; mt < MT; ++mt) {
#pragma unroll
      for (int r = 0; r < 8; ++r) acc[mt][nb][r] = bv;
    }
  }
#pragma unroll 4
  for (int k = 0; k < K; k += 4) {
    v2f a[MT];
#pragma unroll
    for (int mt = 0; mt < MT; ++mt) {
      const float* ar = Alds + (mt * 16 + lm) * astride + k + 2 * lh;
      a[mt].x = ar[0];
      a[mt].y = ar[1];
    }
    const float* wr = W + (k + 2 * lh) * 64 + lm;
#pragma unroll
    for (int nb = 0; nb < 4; ++nb) {
      v2f b;
      b.x = wr[nb * 16];
      b.y = wr[64 + nb * 16];
#pragma unroll
      for (int mt = 0; mt < MT; ++mt)
        acc[mt][nb] = wmma4(a[mt], b, acc[mt][nb]);
    }
  }
}

template<int MT, bool ACT>
__device__ __forceinline__ void frag_to_lds(float* O, int ostride, int lane, v8f acc[MT][4]) {
  const int lm = lane & 15, lh = lane >> 4;
#pragma unroll
  for (int mt = 0; mt < MT; ++mt)
#pragma unroll
    for (int nb = 0; nb < 4; ++nb)
#pragma unroll
      for (int r = 0; r < 8; ++r) {
        float v = acc[mt][nb][r];
        if (ACT) v = silu_f(v);
        O[(mt * 16 + r + 8 * lh) * ostride + nb * 16 + lm] = v;
      }
}

struct alignas(16) EdgeSmem {
  float A[32 * 164];    // e_in (stride 164); reused for m2 (stride 68)
  float M1[32 * 68];
  float diffn[32 * 3];
  float ps[32];
  int   dsts[32];
};

__global__ __launch_bounds__(64, 4) void egnn_edge_kernel(
    const float* __restrict__ h, const float* __restrict__ x,
    const float* __restrict__ edge_attr,
    const int* __restrict__ src, const int* __restrict__ dst,
    const float* __restrict__ ew1p, const float* __restrict__ eb1,
    const float* __restrict__ ew2,  const float* __restrict__ eb2,
    const float* __restrict__ pw1,  const float* __restrict__ pb1,
    const float* __restrict__ pw2,
    float* __restrict__ agg, float* __restrict__ pos_acc, float* __restrict__ cnt,
    int layer) {
  __shared__ EdgeSmem sm[2];
  EdgeSmem& w = sm[threadIdx.x >> 5];
  const int lane   = threadIdx.x & 31;
  const int gwid   = (blockIdx.x * blockDim.x + threadIdx.x) >> 5;
  const int gwaves = (gridDim.x * blockDim.x) >> 5;
  const int ntiles = N_EDGES / 32;   // 25000 exact

  const float* W1 = ew1p + (size_t)layer * KPAD * 64;
  const float* B1 = eb1 + layer * 64;
  const float* W2 = ew2 + (size_t)layer * 64 * 64;
  const float* B2 = eb2 + layer * 64;
  const float* W3 = pw1 + (size_t)layer * 64 * 64;
  const float* B3 = pb1 + layer * 64;
  const float* Wp = pw2 + layer * 64;

  for (int t = gwid; t < ntiles; t += gwaves) {
    const int ebase = t * 32;
    { // ---- gather e_in = [h[dst], h[src], radial, edge_attr, 0-pad] ----
      const int e = ebase + lane;
      const int d = dst[e], s = src[e];
      w.dsts[lane] = d;
      float dx0 = x[d * 3 + 0] - x[s * 3 + 0];
      float dx1 = x[d * 3 + 1] - x[s * 3 + 1];
      float dx2 = x[d * 3 + 2] - x[s * 3 + 2];
      float r2  = dx0 * dx0 + dx1 * dx1 + dx2 * dx2;
      float inv = 1.0f / (sqrtf(r2) + EPS_F);
      w.diffn[lane * 3 + 0] = dx0 * inv;
      w.diffn[lane * 3 + 1] = dx1 * inv;
      w.diffn[lane * 3 + 2] = dx2 * inv;
      float* Ar = w.A + lane * 164;
      const float4* hd = (const float4*)(h + (size_t)d * 64);
      const float4* hs = (const float4*)(h + (size_t)s * 64);
      float4* A4 = (float4*)Ar;
#pragma unroll
      for (int j = 0; j < 16; ++j) A4[j] = hd[j];
#pragma unroll
      for (int j = 0; j < 16; ++j) A4[16 + j] = hs[j];
      Ar[128] = r2;
      const float* ea = edge_attr + (size_t)e * 16;
#pragma unroll
      for (int j = 0; j < 16; ++j) Ar[129 + j] = ea[j];
#pragma unroll
      for (int j = 145; j < 160; ++j) Ar[j] = 0.0f;
    }
    __builtin_amdgcn_wave_barrier();
    v8f acc[2][4];
    // ---- edge MLP: silu(e_in @ W1 + b1) ----
    wave_gemm<2>(w.A, 164, W1, B1, KPAD, lane, acc);
    frag_to_lds<2, true>(w.M1, 68, lane, acc);
    __builtin_amdgcn_wave_barrier();
    // ---- edge MLP: m = silu(m1 @ W2 + b2)  (into A region) ----
    wave_gemm<2>(w.M1, 68, W2, B2, 64, lane, acc);
    frag_to_lds<2, true>(w.A, 68, lane, acc);
    w.ps[lane] = 0.0f;
    __builtin_amdgcn_wave_barrier();
    // ---- pos MLP: ps = silu(m @ W3 + b3) @ pw2 ----
    wave_gemm<2>(w.A, 68, W3, B3, 64, lane, acc);
    {
      const int lm = lane & 15, lh = lane >> 4;
#pragma unroll
      for (int mt = 0; mt < 2; ++mt)
#pragma unroll
        for (int r = 0; r < 8; ++r) {
          float contrib = 0.0f;
#pragma unroll
          for (int nb = 0; nb < 4; ++nb)
            contrib += silu_f(acc[mt][nb][r]) * Wp[nb * 16 + lm];
          atomicAdd(&w.ps[mt * 16 + r + 8 * lh], contrib);
        }
    }
    __builtin_amdgcn_wave_barrier();
    { // ---- scatter: position update terms + message aggregation ----
      const int d    = w.dsts[lane];
      const float pv = w.ps[lane];
      atomicAdd(&pos_acc[d * 3 + 0], w.diffn[lane * 3 + 0] * pv);
      atomicAdd(&pos_acc[d * 3 + 1], w.diffn[lane * 3 + 1] * pv);
      atomicAdd(&pos_acc[d * 3 + 2], w.diffn[lane * 3 + 2] * pv);
      atomicAdd(&cnt[d], 1.0f);
      for (int j = 0; j < 64; ++j) {
        int idx = j * 32 + lane;
        int row = idx >> 6, col = idx & 63;
        atomicAdd(&agg[(size_t)w.dsts[row] * 64 + col], w.A[row * 68 + col]);
      }
    }
    __builtin_amdgcn_wave_barrier();
  }
}

struct alignas(16) NodeSmem {
  float F[32 * 132];  // [h || agg], stride 132
  float T[32 * 68];
};

__global__ __launch_bounds__(64, 4) void egnn_node_kernel(
    float* __restrict__ h, float* __restrict__ x,
    const float* __restrict__ agg, const float* __restrict__ pos_acc,
    const float* __restrict__ cnt,
    const float* __restrict__ fw1, const float* __restrict__ fb1,
    const float* __restrict__ fw2, const float* __restrict__ fb2,
    int layer) {
  __shared__ NodeSmem sm[2];
  NodeSmem& w = sm[threadIdx.x >> 5];
  const int lane   = threadIdx.x & 31;
  const int gwid   = (blockIdx.x * blockDim.x + threadIdx.x) >> 5;
  const int gwaves = (gridDim.x * blockDim.x) >> 5;
  const int ntiles = (N_NODES + 31) / 32;

  const float* W1 = fw1 + (size_t)layer * FEAT_IN * 64;
  const float* B1 = fb1 + layer * 64;
  const float* W2 = fw2 + (size_t)layer * 64 * 64;
  const float* B2 = fb2 + layer * 64;

  for (int t = gwid; t < ntiles; t += gwaves) {
    const int nbase = t * 32;
    {
      const int n = nbase + lane;
      float* Fr = w.F + lane * 132;
      if (n < N_NODES) {
        const float4* hv = (const float4*)(h + (size_t)n * 64);
        const float4* av = (const float4*)(agg + (size_t)n * 64);
        float4* F4 = (float4*)Fr;
#pragma unroll
        for (int j = 0; j < 16; ++j) F4[j] = hv[j];
#pragma unroll
        for (int j = 0; j < 16; ++j) F4[16 + j] = av[j];
        float c = fmaxf(cnt[n], 1.0f);
        x[n * 3 + 0] += pos_acc[n * 3 + 0] / c;
        x[n * 3 + 1] += pos_acc[n * 3 + 1] / c;
        x[n * 3 + 2] += pos_acc[n * 3 + 2] / c;
      } else {
#pragma unroll
        for (int j = 0; j < 128; ++j) Fr[j] = 0.0f;
      }
    }
    __builtin_amdgcn_wave_barrier();
    v8f acc[2][4];
    wave_gemm<2>(w.F, 132, W1, B1, FEAT_IN, lane, acc);
    frag_to_lds<2, true>(w.T, 68, lane, acc);
    __builtin_amdgcn_wave_barrier();
    wave_gemm<2>(w.T, 68, W2, B2, 64, lane, acc);
    {
      const int lm = lane & 15, lh = lane >> 4;
#pragma unroll
      for (int mt = 0; mt < 2; ++mt)
#pragma unroll
        for (int r = 0; r < 8; ++r) {
          int row = mt * 16 + r + 8 * lh;
          int n = nbase + row;
          if (n < N_NODES) {
#pragma unroll
            for (int nb = 0; nb < 4; ++nb) {
              int c = nb * 16 + lm;
              float hv = h[(size_t)n * 64 + c] + acc[mt][nb][r];
              if (layer < DEPTH - 1) hv = silu_f(hv);
              h[(size_t)n * 64 + c] = hv;
            }
          }
        }
    }
    __builtin_amdgcn_wave_barrier();
  }
}

__global__ void egnn_embed_kernel(const int* __restrict__ nf,
                                  const float* __restrict__ emb,
                                  const float* __restrict__ in_w,
                                  const float* __restrict__ in_b,
                                  float* __restrict__ h) {
  __shared__ float er[64];
  const int n = blockIdx.x;
  const int j = threadIdx.x;
  er[j] = emb[(size_t)nf[n] * 64 + j];
  __syncthreads();
  float a = in_b[j];
  for (int k = 0; k < 64; ++k) a += er[k] * in_w[k * 64 + j];
  h[(size_t)n * 64 + j] = a;
}

__global__ void egnn_copy_pos_kernel(const float* __restrict__ p, float* __restrict__ x, int n) {
  int i = blockIdx.x * blockDim.x + threadIdx.x;
  if (i < n) x[i] = p[i];
}

__global__ void egnn_pad_w1_kernel(const float* __restrict__ ew1, float* __restrict__ wp) {
  int i = blockIdx.x * blockDim.x + threadIdx.x;
  if (i >= DEPTH * KPAD * 64) return;
  int l = i / (KPAD * 64);
  int r = (i / 64) % KPAD;
  int c = i % 64;
  wp[i] = (r < EDGE_IN) ? ew1[(size_t)l * EDGE_IN * 64 + r * 64 + c] : 0.0f;
}

__global__ void egnn_zero_kernel(float* __restrict__ p, int n) {
  int i = blockIdx.x * blockDim.x + threadIdx.x;
  if (i < n) p[i] = 0.0f;
}

extern "C" void kernel_launch(void* const* d_in, const int* in_sizes, int n_in,
                              void* d_out, int out_size, void* d_ws, size_t ws_size,
                              hipStream_t stream) {
  const int*   node_feats = (const int*)  d_in[0];
  const float* positions  = (const float*)d_in[1];
  const float* edge_attr  = (const float*)d_in[2];
  const int*   src        = (const int*)  d_in[3];
  const int*   dst        = (const int*)  d_in[4];
  const float* emb_table  = (const float*)d_in[5];
  const float* in_w       = (const float*)d_in[6];
  const float* in_b       = (const float*)d_in[7];
  const float* edge_w1    = (const float*)d_in[8];
  const float* edge_b1    = (const float*)d_in[9];
  const float* edge_w2    = (const float*)d_in[10];
  const float* edge_b2    = (const float*)d_in[11];
  const float* pos_w1     = (const float*)d_in[12];
  const float* pos_b1     = (const float*)d_in[13];
  const float* pos_w2     = (const float*)d_in[14];
  const float* feat_w1    = (const float*)d_in[15];
  const float* feat_b1    = (const float*)d_in[16];
  const float* feat_w2    = (const float*)d_in[17];
  const float* feat_b2    = (const float*)d_in[18];

  float* h = (float*)d_out;                        // [N,64]
  float* x = (float*)d_out + (size_t)N_NODES * 64; // [N,3]

  float* ws      = (float*)d_ws;
  float* agg     = ws;                               // N*64
  float* pos_acc = agg + (size_t)N_NODES * 64;       // N*3
  float* cnt     = pos_acc + (size_t)N_NODES * 3;    // N
  float* ew1p    = cnt + N_NODES;                    // DEPTH*KPAD*64

  egnn_pad_w1_kernel<<<(DEPTH * KPAD * 64 + 255) / 256, 256, 0, stream>>>(edge_w1, ew1p);
  egnn_embed_kernel<<<N_NODES, 64, 0, stream>>>(node_feats, emb_table, in_w, in_b, h);
  egnn_copy_pos_kernel<<<(N_NODES * 3 + 255) / 256, 256, 0, stream>>>(positions, x, N_NODES * 3);

  const int zn = N_NODES * 64 + N_NODES * 3 + N_NODES;
  for (int l = 0; l < DEPTH; ++l) {
    egnn_zero_kernel<<<(zn + 255) / 256, 256, 0, stream>>>(agg, zn);
    egnn_edge_kernel<<<2048, 64, 0, stream>>>(h, x, edge_attr, src, dst,
        ew1p, edge_b1, edge_w2, edge_b2, pos_w1, pos_b1, pos_w2,
        agg, pos_acc, cnt, l);
    egnn_node_kernel<<<512, 64, 0, stream>>>(h, x, agg, pos_acc, cnt,
        feat_w1, feat_b1, feat_w2, feat_b2, l);
  }
}